// MockStreamGenerator_72782515798902
// MI455X (gfx1250) — compile-verified
//
#include <hip/hip_runtime.h>
#include <stdint.h>

#define T_STEPS 2048

// ---------------------------------------------------------------------------
// Softened point-mass acceleration: a = ngm * x * (r^2 + b^2)^{-3/2}
// (ngm = -gm precomputed). b2 folded into the r^2 FMA chain.
// Dependent chain: 3 fma + v_rsq + 2 mul + 1 mul + 1 mul per component.
// ---------------------------------------------------------------------------
__device__ __forceinline__ void accel(float ngm, float b2,
                                      float x, float y, float z,
                                      float& ax, float& ay, float& az) {
  float s   = x * x + y * y + z * z + b2;
  float inv = __builtin_amdgcn_rsqf(s);   // s^-1/2
  float c   = ngm * (inv * inv * inv);    // -gm * s^-3/2
  ax = c * x; ay = c * y; az = c * z;
}

// ---------------------------------------------------------------------------
// Threefry2x32, 20 rounds (counter-based deterministic RNG, threefry family
// like JAX's PRNG; bit-exact JAX match is unverifiable compile-only).
// ---------------------------------------------------------------------------
__device__ __forceinline__ void threefry2x32(uint32_t k0, uint32_t k1,
                                             uint32_t c0, uint32_t c1,
                                             uint32_t& o0, uint32_t& o1) {
  uint32_t ks2 = k0 ^ k1 ^ 0x1BD11BDAu;
  uint32_t x0 = c0 + k0, x1 = c1 + k1;
#define TF_ROUND(r) { x0 += x1; x1 = (x1 << (r)) | (x1 >> (32 - (r))); x1 ^= x0; }
  TF_ROUND(13) TF_ROUND(15) TF_ROUND(26) TF_ROUND(6)
  x0 += k1;  x1 += ks2 + 1u;
  TF_ROUND(17) TF_ROUND(29) TF_ROUND(16) TF_ROUND(24)
  x0 += ks2; x1 += k0 + 2u;
  TF_ROUND(13) TF_ROUND(15) TF_ROUND(26) TF_ROUND(6)
  x0 += k0;  x1 += k1 + 3u;
  TF_ROUND(17) TF_ROUND(29) TF_ROUND(16) TF_ROUND(24)
  x0 += k1;  x1 += ks2 + 4u;
  TF_ROUND(13) TF_ROUND(15) TF_ROUND(26) TF_ROUND(6)
  x0 += ks2; x1 += k0 + 5u;
#undef TF_ROUND
  o0 = x0; o1 = x1;
}

__device__ __forceinline__ float normal_sample(uint32_t k0, uint32_t k1,
                                               uint32_t c0, uint32_t c1) {
  uint32_t a, b;
  threefry2x32(k0, k1, c0, c1, a, b);
  // (0,1) uniforms, then Box-Muller (cosine branch).
  float u1 = ((float)a + 0.5f) * 2.3283064365386963e-10f;
  float u2 = ((float)b + 0.5f) * 2.3283064365386963e-10f;
  float r  = sqrtf(-2.0f * __logf(u1));
  return r * __cosf(6.28318530717958647f * u2);
}

// ---------------------------------------------------------------------------
// Kernel 1: sequential progenitor leapfrog scan (inherently serial). Cached-
// acceleration KDK: one accel eval per step (bit-identical to the reference's
// two-eval form since a(x_{j+1}) is shared between adjacent half-kicks).
// Writes prog_ws into the third output slot; kernel 2 consumes it.
// ---------------------------------------------------------------------------
__global__ void prog_scan_kernel(const float* __restrict__ ts,
                                 const float* __restrict__ w0,
                                 const float* __restrict__ gmp,
                                 const float* __restrict__ bp,
                                 float* __restrict__ out) {
  if (threadIdx.x != 0 || blockIdx.x != 0) return;
  const float gm  = gmp[0];
  const float ngm = -gm;
  const float b0  = bp[0];
  const float b2  = b0 * b0;
  float x  = w0[0], y  = w0[1], z  = w0[2];
  float vx = w0[3], vy = w0[4], vz = w0[5];
  float* prog = out + 2 * T_STEPS * 6;
  prog[0] = x;  prog[1] = y;  prog[2] = z;
  prog[3] = vx; prog[4] = vy; prog[5] = vz;
  float ax, ay, az;
  accel(ngm, b2, x, y, z, ax, ay, az);
  float tprev = ts[0];
  for (int j = 1; j < T_STEPS; ++j) {
    float tcur = ts[j];
    float dt = tcur - tprev;
    tprev = tcur;
    float h = 0.5f * dt;
    vx += h * ax; vy += h * ay; vz += h * az;   // half kick (cached a)
    x += dt * vx; y += dt * vy; z += dt * vz;   // drift
    accel(ngm, b2, x, y, z, ax, ay, az);        // single eval per step
    vx += h * ax; vy += h * ay; vz += h * az;   // half kick
    float* o = prog + j * 6;
    o[0] = x;  o[1] = y;  o[2] = z;
    o[3] = vx; o[4] = vy; o[5] = vz;
  }
}

// ---------------------------------------------------------------------------
// Kernel 2: one thread per stream particle (2048 lead + 2048 trail).
// dts[] staged per block into LDS via the CDNA5 async global->LDS path
// (ASYNCcnt-tracked) so the serial inner loop touches only LDS + VALU.
// ---------------------------------------------------------------------------
__global__ void __launch_bounds__(128)
stream_kernel(const float* __restrict__ ts,
              const float* __restrict__ pmassp,
              const float* __restrict__ gmp,
              const float* __restrict__ bp,
              const int* __restrict__ seedp,
              float* __restrict__ out) {
  __shared__ __align__(16) float s_ts[T_STEPS];
  __shared__ __align__(16) float s_dts[T_STEPS];  // [0 .. T-2] used

  const int tid = threadIdx.x;

  // --- Async-stage ts[0..2047] (8 KB) into LDS; tracked by ASYNCcnt. ---
  {
    uint32_t lds_base = (uint32_t)(uintptr_t)(&s_ts[0]); // low 32b of flat = LDS offset
#pragma unroll
    for (int i = 0; i < T_STEPS / 128; ++i) {
      int e = tid + i * 128;
      uint64_t ga = (uint64_t)(uintptr_t)(ts + e);
      uint32_t la = lds_base + (uint32_t)e * 4u;
      asm volatile("global_load_async_to_lds_b32 %0, %1, off"
                   :: "v"(la), "v"(ga) : "memory");
    }
    asm volatile("s_wait_asynccnt 0x0" ::: "memory");
  }
  __syncthreads();
  for (int e = tid; e < T_STEPS - 1; e += 128) s_dts[e] = s_ts[e + 1] - s_ts[e];
  __syncthreads();

  const int gid   = blockIdx.x * 128 + tid;   // 0..4095
  const int p     = gid & (T_STEPS - 1);      // release step; trip count T-1-p differs per lane
  const int trail = gid >> 11;                // 0 = lead, 1 = trail

  const float gm  = gmp[0];
  const float ngm = -gm;
  const float b0  = bp[0];
  const float b2  = b0 * b0;
  const float pm  = pmassp[0];
  const uint32_t seed = (uint32_t)seedp[0];

  // Progenitor state at release time (written by kernel 1 into out[2T*6..)).
  const float* pr = out + 2 * T_STEPS * 6 + p * 6;
  const float px  = pr[0], py  = pr[1], pz  = pr[2];
  const float pvx = pr[3], pvy = pr[4], pvz = pr[5];

  const float r     = sqrtf(px * px + py * py + pz * pz);
  const float rinv  = 1.0f / r;
  const float rt    = r * cbrtf(pm / (3.0f * gm));
  const float sigma = sqrtf(pm / (rt + 1e-8f));
  const float sgn   = trail ? 1.0f : -1.0f;       // lead: -rt*rhat, trail: +rt*rhat
  const uint32_t kpos = trail ? 2u : 1u;          // k2 : k1
  const uint32_t kvel = trail ? 4u : 3u;          // k4 : k3

  const float n0 = normal_sample(seed, kpos, (uint32_t)p, 0u);
  const float n1 = normal_sample(seed, kpos, (uint32_t)p, 1u);
  const float n2 = normal_sample(seed, kpos, (uint32_t)p, 2u);
  const float m0 = normal_sample(seed, kvel, (uint32_t)p, 0u);
  const float m1 = normal_sample(seed, kvel, (uint32_t)p, 1u);
  const float m2 = normal_sample(seed, kvel, (uint32_t)p, 2u);

  const float POS_SCATTER = 0.25f, VEL_SCATTER = 0.3f;
  float x = px + sgn * rt * px * rinv + POS_SCATTER * rt * n0;
  float y = py + sgn * rt * py * rinv + POS_SCATTER * rt * n1;
  float z = pz + sgn * rt * pz * rinv + POS_SCATTER * rt * n2;
  float vx = pvx + VEL_SCATTER * sigma * m0;
  float vy = pvy + VEL_SCATTER * sigma * m1;
  float vz = pvz + VEL_SCATTER * sigma * m2;

  // Serial leapfrog from release step p. Cached-acceleration KDK: one accel
  // eval per step -> ~11 dependent VALU ops/step (was ~22). dt comes from LDS.
  float ax, ay, az;
  accel(ngm, b2, x, y, z, ax, ay, az);
  for (int j = p; j < T_STEPS - 1; ++j) {
    float dt = s_dts[j];
    float h  = 0.5f * dt;
    vx += h * ax; vy += h * ay; vz += h * az;   // half kick (cached a)
    x += dt * vx; y += dt * vy; z += dt * vz;   // drift
    accel(ngm, b2, x, y, z, ax, ay, az);        // single eval per step
    vx += h * ax; vy += h * ay; vz += h * az;   // half kick
  }

  float* o = out + trail * T_STEPS * 6 + p * 6;
  o[0] = x;  o[1] = y;  o[2] = z;
  o[3] = vx; o[4] = vy; o[5] = vz;
}

// ---------------------------------------------------------------------------
// Launch: kernel 1 (serial scan) then kernel 2 (4096 particle threads).
// ---------------------------------------------------------------------------
extern "C" void kernel_launch(void* const* d_in, const int* in_sizes, int n_in,
                              void* d_out, int out_size, void* d_ws, size_t ws_size,
                              hipStream_t stream) {
  (void)in_sizes; (void)n_in; (void)out_size; (void)d_ws; (void)ws_size;
  const float* ts    = (const float*)d_in[0];
  const float* w0    = (const float*)d_in[1];
  const float* pmass = (const float*)d_in[2];
  const float* gm    = (const float*)d_in[3];
  const float* b     = (const float*)d_in[4];
  const int*   seed  = (const int*)d_in[5];
  float* out = (float*)d_out;

  prog_scan_kernel<<<1, 32, 0, stream>>>(ts, w0, gm, b, out);
  stream_kernel<<<(2 * T_STEPS) / 128, 128, 0, stream>>>(ts, pmass, gm, b, seed, out);
}